// SimpleGAT_55645596287280
// MI455X (gfx1250) — compile-verified
//
#include <hip/hip_runtime.h>
#include <math.h>

#define N_NODES 20000
#define N_EDGES 320000
#define IN_CH 512
#define HID 128
#define HEADS 4
#define OUT_CH 256
#define NEG_SLOPE 0.2f

typedef float v2f __attribute__((ext_vector_type(2)));
typedef float v8f __attribute__((ext_vector_type(8)));

// ---------------- utility: ordered-int encoding for float atomic max --------
__device__ __forceinline__ int ford(float f) {
    int i = __float_as_int(f);
    return (i >= 0) ? i : (i ^ 0x7FFFFFFF);
}
__device__ __forceinline__ float iord(int i) {
    return __int_as_float((i >= 0) ? i : (i ^ 0x7FFFFFFF));
}

// ---------------- fill kernels ----------------------------------------------
__global__ void fill_f32(float* __restrict__ p, float v, int n) {
    int i = blockIdx.x * blockDim.x + threadIdx.x;
    if (i < n) p[i] = v;
}
__global__ void fill_i32(int* __restrict__ p, int v, int n) {
    int i = blockIdx.x * blockDim.x + threadIdx.x;
    if (i < n) p[i] = v;
}

// ---------------- WMMA f32 GEMM: C[M x NOUT] = A[M x K] @ B[K x NOUT] -------
// One wave per block. Each wave owns a 16 x 64 output strip:
//   blockIdx.x -> M tile (16 rows), blockIdx.y -> N group (4 x 16 cols).
// A (16x4 f32) layout: lane l: m = l&15, kb = (l>>4)*2, vgpr0=K=kb, vgpr1=K=kb+1
// B (4x16 f32) layout: lane l: n = l&15, kb = (l>>4)*2, vgpr0=K=kb, vgpr1=K=kb+1
// C/D (16x16 f32): vgpr r: M = r + 8*(l>>4), N = l&15
template <int K, int NOUT>
__global__ __launch_bounds__(32)
void gemm_wmma_f32(const float* __restrict__ A, const float* __restrict__ B,
                   float* __restrict__ C) {
    const int lane = threadIdx.x & 31;
    const int m0 = blockIdx.x * 16;
    const int n0 = blockIdx.y * 64;
    const int mr = lane & 15;
    const int nr = lane & 15;
    const int kb = (lane >> 4) * 2;

    v8f acc0 = {}, acc1 = {}, acc2 = {}, acc3 = {};
    const float* __restrict__ Arow = A + (size_t)(m0 + mr) * K + kb;

    for (int k = 0; k < K; k += 4) {
        v2f a;
        a.x = Arow[k + 0];
        a.y = Arow[k + 1];

        const float* __restrict__ Bp = B + (size_t)(k + kb) * NOUT + n0 + nr;
        v2f b0, b1, b2, b3;
        b0.x = Bp[0];      b0.y = Bp[NOUT];
        b1.x = Bp[16];     b1.y = Bp[NOUT + 16];
        b2.x = Bp[32];     b2.y = Bp[NOUT + 32];
        b3.x = Bp[48];     b3.y = Bp[NOUT + 48];

        acc0 = __builtin_amdgcn_wmma_f32_16x16x4_f32(false, a, false, b0, (short)0, acc0, false, false);
        acc1 = __builtin_amdgcn_wmma_f32_16x16x4_f32(false, a, false, b1, (short)0, acc1, false, false);
        acc2 = __builtin_amdgcn_wmma_f32_16x16x4_f32(false, a, false, b2, (short)0, acc2, false, false);
        acc3 = __builtin_amdgcn_wmma_f32_16x16x4_f32(false, a, false, b3, (short)0, acc3, false, false);
    }

    const int half = lane >> 4;
#pragma unroll
    for (int r = 0; r < 8; ++r) {
        float* __restrict__ Cp = C + (size_t)(m0 + r + 8 * half) * NOUT + n0 + nr;
        Cp[0]  = acc0[r];
        Cp[16] = acc1[r];
        Cp[32] = acc2[r];
        Cp[48] = acc3[r];
    }
}

// ---------------- attention dot products: a_s[n,h] = <H[n,h,:], a_src[h,:]> --
// One wave per (node, head).
template <int CH>
__global__ __launch_bounds__(256)
void att_dots(const float* __restrict__ Hm, const float* __restrict__ a_src,
              const float* __restrict__ a_dst, int nwaves, int heads,
              float* __restrict__ as_out, float* __restrict__ ad_out) {
    int gid = blockIdx.x * blockDim.x + threadIdx.x;
    int wave = gid >> 5;
    int lane = gid & 31;
    if (wave >= nwaves) return;
    int h = wave % heads;
    const float* __restrict__ hp  = Hm + (size_t)wave * CH;   // contiguous [n][h][CH]
    const float* __restrict__ asv = a_src + h * CH;
    const float* __restrict__ adv = a_dst + h * CH;
    float s = 0.f, d = 0.f;
#pragma unroll
    for (int c = lane; c < CH; c += 32) {
        float v = hp[c];
        s += v * asv[c];
        d += v * adv[c];
    }
#pragma unroll
    for (int off = 16; off > 0; off >>= 1) {
        s += __shfl_xor(s, off, 32);
        d += __shfl_xor(d, off, 32);
    }
    if (lane == 0) { as_out[wave] = s; ad_out[wave] = d; }
}

// ---------------- edge pass 1: logits + segment max --------------------------
template <int H>
__global__ __launch_bounds__(256)
void edge_logits(const int* __restrict__ ei, const float* __restrict__ as,
                 const float* __restrict__ ad, float* __restrict__ elog,
                 int* __restrict__ emax) {
    int idx = blockIdx.x * blockDim.x + threadIdx.x;
    const int total = (N_EDGES + N_NODES) * H;
    if (idx >= total) return;
    int e = idx / H, h = idx % H;
    int s = (e < N_EDGES) ? ei[e] : (e - N_EDGES);
    int d = (e < N_EDGES) ? ei[N_EDGES + e] : (e - N_EDGES);
    float v = as[s * H + h] + ad[d * H + h];
    v = (v > 0.f) ? v : NEG_SLOPE * v;
    elog[idx] = v;
    atomicMax(&emax[d * H + h], ford(v));
}

// ---------------- edge pass 2: exp(e - max) + segment sum --------------------
template <int H>
__global__ __launch_bounds__(256)
void edge_exp(const int* __restrict__ ei, float* __restrict__ elog,
              const int* __restrict__ emax, float* __restrict__ denom) {
    int idx = blockIdx.x * blockDim.x + threadIdx.x;
    const int total = (N_EDGES + N_NODES) * H;
    if (idx >= total) return;
    int e = idx / H, h = idx % H;
    int d = (e < N_EDGES) ? ei[N_EDGES + e] : (e - N_EDGES);
    float m = iord(emax[d * H + h]);
    float ex = expf(elog[idx] - m);
    elog[idx] = ex;
    atomicAdd(&denom[d * H + h], ex);
}

// ---------------- edge pass 3: normalize, emit alpha -------------------------
template <int H>
__global__ __launch_bounds__(256)
void edge_alpha(const int* __restrict__ ei, float* __restrict__ elog,
                const float* __restrict__ denom, float* __restrict__ alpha_out) {
    int idx = blockIdx.x * blockDim.x + threadIdx.x;
    const int total = (N_EDGES + N_NODES) * H;
    if (idx >= total) return;
    int e = idx / H, h = idx % H;
    int d = (e < N_EDGES) ? ei[N_EDGES + e] : (e - N_EDGES);
    float a = elog[idx] / denom[d * H + h];
    elog[idx] = a;
    alpha_out[idx] = a;
}

// ---------------- edge pass 4: weighted scatter-add of messages --------------
// One wave per (edge, head); CH channels strided across lanes.
template <int H, int CH>
__global__ __launch_bounds__(256)
void edge_msg(const int* __restrict__ ei, const float* __restrict__ Hm,
              const float* __restrict__ alpha, float* __restrict__ agg) {
    int gid = blockIdx.x * blockDim.x + threadIdx.x;
    int wave = gid >> 5;
    int lane = gid & 31;
    const int total = (N_EDGES + N_NODES) * H;
    if (wave >= total) return;
    int e = wave / H, h = wave % H;
    int s = (e < N_EDGES) ? ei[e] : (e - N_EDGES);
    int d = (e < N_EDGES) ? ei[N_EDGES + e] : (e - N_EDGES);
    float al = alpha[wave];
    const float* __restrict__ hp = Hm + (size_t)s * (H * CH) + h * CH;
    float* __restrict__ op = agg + (size_t)d * (H * CH) + h * CH;
#pragma unroll
    for (int c = lane; c < CH; c += 32)
        atomicAdd(&op[c], al * hp[c]);
}

// ---------------- ELU(x + bias) in place -------------------------------------
template <int CH>
__global__ __launch_bounds__(256)
void elu_bias_inplace(float* __restrict__ p, const float* __restrict__ bias, int n) {
    int i = blockIdx.x * blockDim.x + threadIdx.x;
    if (i >= n) return;
    float v = p[i] + bias[i & (CH - 1)];
    p[i] = (v > 0.f) ? v : (expf(v) - 1.f);
}

// ---------------- out += bias in place ---------------------------------------
template <int CH>
__global__ __launch_bounds__(256)
void bias_inplace(float* __restrict__ p, const float* __restrict__ bias, int n) {
    int i = blockIdx.x * blockDim.x + threadIdx.x;
    if (i >= n) return;
    p[i] += bias[i & (CH - 1)];
}

static inline int cdiv(long long a, int b) { return (int)((a + b - 1) / b); }

extern "C" void kernel_launch(void* const* d_in, const int* in_sizes, int n_in,
                              void* d_out, int out_size, void* d_ws, size_t ws_size,
                              hipStream_t stream) {
    const float* x      = (const float*)d_in[0];
    const int*   ei     = (const int*)d_in[1];
    const float* W1     = (const float*)d_in[2];
    const float* a_src1 = (const float*)d_in[3];
    const float* a_dst1 = (const float*)d_in[4];
    const float* b1     = (const float*)d_in[5];
    const float* W2     = (const float*)d_in[6];
    const float* a_src2 = (const float*)d_in[7];
    const float* a_dst2 = (const float*)d_in[8];
    const float* b2     = (const float*)d_in[9];

    const int EP = N_EDGES + N_NODES;  // edges + self loops

    // ---- workspace carve-up (floats) ----
    float* H1   = (float*)d_ws;                          // [N, 512] layer-1 features
    float* AGG1 = H1   + (size_t)N_NODES * 512;          // [N, 512] layer-1 agg -> h2
    float* AS1  = AGG1 + (size_t)N_NODES * 512;          // [N, 4]
    float* AD1  = AS1  + (size_t)N_NODES * HEADS;
    float* DEN1 = AD1  + (size_t)N_NODES * HEADS;
    int*   EMX1 = (int*)(DEN1 + (size_t)N_NODES * HEADS);
    float* ELG1 = (float*)(EMX1 + (size_t)N_NODES * HEADS);  // [E', 4]
    float* AS2  = ELG1 + (size_t)EP * HEADS;
    float* AD2  = AS2  + N_NODES;
    float* DEN2 = AD2  + N_NODES;
    int*   EMX2 = (int*)(DEN2 + N_NODES);
    float* ELG2 = (float*)(EMX2 + N_NODES);              // [E', 1]
    float* G2   = H1;  // alias: H1 is dead after layer-1 messages complete

    // ---- output carve-up ----
    float* out    = (float*)d_out;                            // [N, 256]
    float* alpha1 = out    + (size_t)N_NODES * OUT_CH;        // [E', 4]
    float* alpha2 = alpha1 + (size_t)EP * HEADS;              // [E', 1]

    const int TB = 256;

    // ===== Layer 1 =====
    // GEMM1: H1 = X @ W1   [20000x512] @ [512x512]
    gemm_wmma_f32<IN_CH, HEADS * HID>
        <<<dim3(N_NODES / 16, (HEADS * HID) / 64), 32, 0, stream>>>(x, W1, H1);

    // attention dot products
    att_dots<HID><<<cdiv((long long)N_NODES * HEADS * 32, TB), TB, 0, stream>>>(
        H1, a_src1, a_dst1, N_NODES * HEADS, HEADS, AS1, AD1);

    // init accumulators
    fill_f32<<<cdiv((long long)N_NODES * 512, TB), TB, 0, stream>>>(AGG1, 0.f, N_NODES * 512);
    fill_f32<<<cdiv((long long)N_NODES * HEADS, TB), TB, 0, stream>>>(DEN1, 0.f, N_NODES * HEADS);
    fill_i32<<<cdiv((long long)N_NODES * HEADS, TB), TB, 0, stream>>>(EMX1, (int)0x80000000, N_NODES * HEADS);

    // edge passes
    edge_logits<HEADS><<<cdiv((long long)EP * HEADS, TB), TB, 0, stream>>>(ei, AS1, AD1, ELG1, EMX1);
    edge_exp<HEADS><<<cdiv((long long)EP * HEADS, TB), TB, 0, stream>>>(ei, ELG1, EMX1, DEN1);
    edge_alpha<HEADS><<<cdiv((long long)EP * HEADS, TB), TB, 0, stream>>>(ei, ELG1, DEN1, alpha1);
    edge_msg<HEADS, HID><<<cdiv((long long)EP * HEADS * 32, TB), TB, 0, stream>>>(ei, H1, ELG1, AGG1);

    // h2 = ELU(agg + b1), in place
    elu_bias_inplace<512><<<cdiv((long long)N_NODES * 512, TB), TB, 0, stream>>>(AGG1, b1, N_NODES * 512);

    // ===== Layer 2 =====
    // GEMM2: G2 = h2 @ W2   [20000x512] @ [512x256]  (G2 aliases H1)
    gemm_wmma_f32<HEADS * HID, OUT_CH>
        <<<dim3(N_NODES / 16, OUT_CH / 64), 32, 0, stream>>>(AGG1, W2, G2);

    att_dots<OUT_CH><<<cdiv((long long)N_NODES * 32, TB), TB, 0, stream>>>(
        G2, a_src2, a_dst2, N_NODES, 1, AS2, AD2);

    fill_f32<<<cdiv((long long)N_NODES * OUT_CH, TB), TB, 0, stream>>>(out, 0.f, N_NODES * OUT_CH);
    fill_f32<<<cdiv((long long)N_NODES, TB), TB, 0, stream>>>(DEN2, 0.f, N_NODES);
    fill_i32<<<cdiv((long long)N_NODES, TB), TB, 0, stream>>>(EMX2, (int)0x80000000, N_NODES);

    edge_logits<1><<<cdiv((long long)EP, TB), TB, 0, stream>>>(ei, AS2, AD2, ELG2, EMX2);
    edge_exp<1><<<cdiv((long long)EP, TB), TB, 0, stream>>>(ei, ELG2, EMX2, DEN2);
    edge_alpha<1><<<cdiv((long long)EP, TB), TB, 0, stream>>>(ei, ELG2, DEN2, alpha2);
    edge_msg<1, OUT_CH><<<cdiv((long long)EP * 32, TB), TB, 0, stream>>>(ei, G2, ELG2, out);

    bias_inplace<OUT_CH><<<cdiv((long long)N_NODES * OUT_CH, TB), TB, 0, stream>>>(out, b2, N_NODES * OUT_CH);
}